// Data_augV4_2173253452141
// MI455X (gfx1250) — compile-verified
//
#include <hip/hip_runtime.h>
#include <hip/hip_bf16.h>

typedef __attribute__((ext_vector_type(2))) float v2f;
typedef __attribute__((ext_vector_type(8))) float v8f;

#define NBATCH     128
#define WIDTH      224
#define CH_ELEMS   50176      // 224*224
#define IMG_ELEMS  150528     // 3*224*224
#define IMG_F4     37632      // IMG_ELEMS/4
#define BPI        147        // blocks per image (256 float4 per block)
#define NBLOCKS    (NBATCH * BPI)   // 18816

// ---- pointwise transform (flips/identity are handled as index remaps) ----
__device__ __forceinline__ float tf_apply(float v, int tf, float m) {
    switch (tf) {
        case 3: return fminf(fmaxf(v * 1.5f, 0.0f), 1.0f);                 // brightness
        case 4: return fminf(fmaxf((v - m) * 1.5f + m, 0.0f), 1.0f);       // contrast
        case 5: return (v < 0.5f) ? v : 1.0f - v;                          // solarize
        case 6: return 1.0f - v;                                           // invert
        case 7: return floorf(v * 4.0f) * 0.25f;                           // posterize
        default: return v;                                                 // 0,1,2: identity/flips
    }
}

// ---- wave32 sum via V_WMMA_F32_16X16X4_F32 with B = ones --------------------
// D[m][n] = sum_k A[m][k]; summing all 16 row-sums == sum of all A entries,
// independent of A's lane layout. D layout: VGPR r -> M=r (lanes 0-15) /
// M=r+8 (lanes 16-31), replicated across N since B is all ones.
__device__ __forceinline__ float wave_sum_wmma(float p) {
    v2f a; a[0] = p;    a[1] = 0.0f;
    v2f b; b[0] = 1.0f; b[1] = 1.0f;
    v8f c = {};
    v8f d = __builtin_amdgcn_wmma_f32_16x16x4_f32(
        /*neg_a=*/false, a, /*neg_b=*/false, b,
        /*c_mod=*/(short)0, c, /*reuse_a=*/false, /*reuse_b=*/false);
    float s = d[0] + d[1] + d[2] + d[3] + d[4] + d[5] + d[6] + d[7];
    s += __shfl_xor(s, 16, 32);   // combine the two half-wave row groups
    return s;                     // every lane now holds the wave total
}

__device__ __forceinline__ void block_reduce_store(float p, float* dst) {
    __shared__ float smem[8];     // 256 threads = 8 waves
    float s = wave_sum_wmma(p);
    int tid = threadIdx.x;
    if ((tid & 31) == 0) smem[tid >> 5] = s;
    __syncthreads();
    if (tid == 0) {
        float t = 0.0f;
#pragma unroll
        for (int i = 0; i < 8; ++i) t += smem[i];
        *dst = t;
    }
}

// ---- pass A: per-block partial sums of x -----------------------------------
__global__ void k_reduceA(const float* __restrict__ x, float* __restrict__ partials) {
    int g = blockIdx.x * 256 + threadIdx.x;      // float4 index
    float4 v = ((const float4*)x)[g];
    block_reduce_store(v.x + v.y + v.z + v.w, &partials[blockIdx.x]);
}

__global__ void k_finalA(const float* __restrict__ partials, float* __restrict__ mean0) {
    int b = threadIdx.x;                          // 128 threads
    double s = 0.0;
    for (int i = 0; i < BPI; ++i) s += (double)partials[b * BPI + i];
    mean0[b] = (float)(s * (1.0 / (double)IMG_ELEMS));
}

// ---- pass B: partial sums of T1(x); skip when mean is analytic -------------
__global__ void k_reduceB(const float* __restrict__ x, const int* __restrict__ samples,
                          const float* __restrict__ mean0, float* __restrict__ partials) {
    int b = blockIdx.x / BPI;
    int tf1 = samples[b];                         // step-0 transform
    if (tf1 == 0 || tf1 == 1 || tf1 == 2 || tf1 == 6) return;  // block-uniform exit
    float m0 = mean0[b];
    int g = blockIdx.x * 256 + threadIdx.x;
    float4 v = ((const float4*)x)[g];
    float p = tf_apply(v.x, tf1, m0) + tf_apply(v.y, tf1, m0) +
              tf_apply(v.z, tf1, m0) + tf_apply(v.w, tf1, m0);
    block_reduce_store(p, &partials[blockIdx.x]);
}

__global__ void k_finalB(const float* __restrict__ partials, const int* __restrict__ samples,
                         const float* __restrict__ mean0, float* __restrict__ mean1) {
    int b = threadIdx.x;                          // 128 threads
    int tf1 = samples[b];
    float m0 = mean0[b];
    float m1;
    if (tf1 == 0 || tf1 == 1 || tf1 == 2) {
        m1 = m0;                                  // flips/identity preserve mean
    } else if (tf1 == 6) {
        m1 = 1.0f - m0;                           // invert: exact
    } else {
        double s = 0.0;
        for (int i = 0; i < BPI; ++i) s += (double)partials[b * BPI + i];
        m1 = (float)(s * (1.0 / (double)IMG_ELEMS));
    }
    mean1[b] = m1;
}

// ---- pass C: out = g2(g1(x[composed-flip indices])) ------------------------
__global__ void k_apply(const float* __restrict__ x, const int* __restrict__ samples,
                        const float* __restrict__ mean0, const float* __restrict__ mean1,
                        float* __restrict__ out) {
    int b   = blockIdx.x / BPI;
    int rem = blockIdx.x % BPI;
    int tf1 = samples[b];
    int tf2 = samples[NBATCH + b];
    float m0 = mean0[b];
    float m1 = mean1[b];
    bool fh = (tf1 == 2) != (tf2 == 2);           // flip_ud XOR
    bool fw = (tf1 == 1) != (tf2 == 1);           // flip_lr XOR

    int local = (rem * 256 + threadIdx.x) * 4;    // element index within image
    int c = local / CH_ELEMS;
    int r = local - c * CH_ELEMS;
    int h = r / WIDTH;
    int w = r - h * WIDTH;                        // multiple of 4
    int hs = fh ? (223 - h) : h;

    const float* src = x + (size_t)b * IMG_ELEMS + (size_t)c * CH_ELEMS + (size_t)hs * WIDTH;
    float4 v;
    if (fw) {
        // out elems w..w+3 come from src elems 223-w..220-w (reversed float4)
        float4 t = *(const float4*)(src + (220 - w));
        v.x = t.w; v.y = t.z; v.z = t.y; v.w = t.x;
    } else {
        v = *(const float4*)(src + w);
    }
    v.x = tf_apply(tf_apply(v.x, tf1, m0), tf2, m1);
    v.y = tf_apply(tf_apply(v.y, tf1, m0), tf2, m1);
    v.z = tf_apply(tf_apply(v.z, tf1, m0), tf2, m1);
    v.w = tf_apply(tf_apply(v.w, tf1, m0), tf2, m1);

    ((float4*)out)[(size_t)blockIdx.x * 256 + threadIdx.x] = v;
}

extern "C" void kernel_launch(void* const* d_in, const int* in_sizes, int n_in,
                              void* d_out, int out_size, void* d_ws, size_t ws_size,
                              hipStream_t stream) {
    const float* x       = (const float*)d_in[0];
    // d_in[1] = prob: unused (dead code in the reference)
    const int*   samples = (const int*)d_in[2];   // [0..127]=step0, [128..255]=step1
    float*       out     = (float*)d_out;

    float* wsf   = (float*)d_ws;                  // ~152 KB used
    float* mean0 = wsf;                           // 128
    float* mean1 = wsf + NBATCH;                  // 128
    float* pA    = wsf + 2 * NBATCH;              // NBLOCKS
    float* pB    = pA + NBLOCKS;                  // NBLOCKS

    k_reduceA<<<NBLOCKS, 256, 0, stream>>>(x, pA);
    k_finalA <<<1,      128, 0, stream>>>(pA, mean0);
    k_reduceB<<<NBLOCKS, 256, 0, stream>>>(x, samples, mean0, pB);
    k_finalB <<<1,      128, 0, stream>>>(pB, samples, mean0, mean1);
    k_apply  <<<NBLOCKS, 256, 0, stream>>>(x, samples, mean0, mean1, out);
}